// LightweightGraphTransformer_55662776156344
// MI455X (gfx1250) — compile-verified
//
#include <hip/hip_runtime.h>
#include <math.h>

// ---------------- problem constants (match reference) ----------------
#define N_NODES     20000
#define N_EDGES     320000
#define D_INF       256
#define DM          128       // hidden dim
#define NHEAD       4
#define HD          512       // NHEAD * DM
#define NLAYER      2
#define NUM_CLASSES 7
#define LN_EPS      1e-5f

typedef __attribute__((ext_vector_type(2))) float v2f;
typedef __attribute__((ext_vector_type(8))) float v8f;
typedef __attribute__((ext_vector_type(4))) unsigned int u32x4;
typedef __attribute__((ext_vector_type(8))) int i32x8;
typedef __attribute__((ext_vector_type(4))) int i32x4;

#if defined(__HIP_DEVICE_COMPILE__) && __has_builtin(__builtin_amdgcn_tensor_load_to_lds)
#define HAVE_TDM 1
#else
#define HAVE_TDM 0
#endif

#if defined(__clang_major__) && (__clang_major__ >= 23)
#define TDM_6ARG 1
#else
#define TDM_6ARG 0
#endif

// =====================================================================
// TDM: DMA a (K x 64) f32 tile of row-major B[K x Nout] (col offset tn)
// into LDS offset 0.  If pad!=0, insert 1 DWORD after every 64 DWORDs
// so LDS row stride becomes 65 floats (kills the lane L / L+16 bank
// conflict on the WMMA B-fragment reads).
// D# layout per CDNA5 ISA ch.8 (group0 128b, group1 256b).
// =====================================================================
#if HAVE_TDM
__device__ __forceinline__ void tdm_load_B_panel(const float* Bsrc, int tn,
                                                 int K, int Nout, int pad)
{
    unsigned long long ga = (unsigned long long)(uintptr_t)(Bsrc + tn);

    u32x4 g0;
    g0[0] = 1u;                                            // count=1, user D#
    g0[1] = 0u;                                            // lds_addr = 0 (dyn-LDS base)
    g0[2] = (unsigned)(ga & 0xFFFFFFFFull);                // global_addr[31:0]
    g0[3] = (unsigned)((ga >> 32) & 0x1FFFFFFull) | (2u << 30);  // addr[56:32] | type=2

    unsigned w0 = (2u << 16);                              // data_size = 4 bytes
    if (pad) w0 |= (1u << 20) | (5u << 22);                // pad_enable, every 64 DW, +1 DW
    i32x8 g1;
    g1[0] = (int)w0;
    g1[1] = (int)(((unsigned)Nout & 0xFFFFu) << 16);       // tensor_dim0 lo16
    g1[2] = (int)((((unsigned)Nout >> 16) & 0xFFFFu) |
                  (((unsigned)K & 0xFFFFu) << 16));        // dim0 hi16 | tensor_dim1 lo16
    g1[3] = (int)((((unsigned)K >> 16) & 0xFFFFu) |
                  (64u << 16));                            // dim1 hi16 | tile_dim0 = 64
    g1[4] = (int)((unsigned)K & 0xFFFFu);                  // tile_dim1 = K, tile_dim2 = 0
    g1[5] = (int)(unsigned)Nout;                           // tensor_dim0_stride lo32
    g1[6] = 0;                                             // stride hi16 | dim1_stride lo16
    g1[7] = 0;

    i32x4 z4 = {0, 0, 0, 0};
#if TDM_6ARG
    i32x8 z8 = {0, 0, 0, 0, 0, 0, 0, 0};
    __builtin_amdgcn_tensor_load_to_lds(g0, g1, z4, z4, z8, 0);
#else
    __builtin_amdgcn_tensor_load_to_lds(g0, g1, z4, z4, 0);
#endif
    __builtin_amdgcn_s_wait_tensorcnt(0);
}
#endif

// =====================================================================
// WMMA f32 GEMM:  C[M x Nout] = A[M x K] @ B[K x Nout] + bias[Nout]
// Block = 8 waves sharing one 64-column group; B panel staged in LDS
// (TDM async DMA), each wave computes a 16x64 strip (4 accumulators).
// V_WMMA_F32_16X16X4_F32 fragment layouts per ISA 7.12.2.
// bstride = 64 (no pad) or 65 (padded rows).
// =====================================================================
__global__ void gemm_wmma_f32(const float* __restrict__ A,
                              const float* __restrict__ B,
                              const float* __restrict__ bias,
                              float* __restrict__ C,
                              int M, int K, int Nout, int bstride)
{
    extern __shared__ float Bs[];                   // K * bstride floats
    const int lane   = threadIdx.x & 31;
    const int waveId = threadIdx.x >> 5;
    const int nGroups = Nout >> 6;
    const int Mtiles  = (M + 15) >> 4;
    const int blkRow  = blockIdx.x / nGroups;
    const int tn      = (blockIdx.x - blkRow * nGroups) << 6;
    const int tm      = blkRow * 8 + waveId;

    // ---- stage B panel (K x 64) into LDS ----
#if HAVE_TDM
    if (waveId == 0)
        tdm_load_B_panel(B, tn, K, Nout, bstride - 64);
#else
    for (int idx = threadIdx.x; idx < K * 64; idx += blockDim.x) {
        int kr = idx >> 6, c = idx & 63;
        Bs[kr * bstride + c] = B[(size_t)kr * Nout + tn + c];
    }
#endif
    __syncthreads();

    if (tm >= Mtiles) return;

    const int row   = tm * 16 + (lane & 15);
    const int c     = lane & 15;
    const int khalf = (lane >> 4) << 1;             // 0 or 2
    const float* __restrict__ Arow = A + (size_t)row * K;

    v8f acc0 = {}, acc1 = {}, acc2 = {}, acc3 = {};
    for (int k0 = 0; k0 < K; k0 += 4) {
        const int ka = k0 + khalf;
        v2f a;
        a.x = Arow[ka];
        a.y = Arow[ka + 1];
        const float* Br0 = Bs + ka * bstride + c;
        const float* Br1 = Br0 + bstride;
        v2f b0, b1, b2, b3;
        b0.x = Br0[0];   b0.y = Br1[0];
        b1.x = Br0[16];  b1.y = Br1[16];
        b2.x = Br0[32];  b2.y = Br1[32];
        b3.x = Br0[48];  b3.y = Br1[48];
        acc0 = __builtin_amdgcn_wmma_f32_16x16x4_f32(false, a, false, b0, (short)0, acc0, false, false);
        acc1 = __builtin_amdgcn_wmma_f32_16x16x4_f32(false, a, false, b1, (short)0, acc1, false, false);
        acc2 = __builtin_amdgcn_wmma_f32_16x16x4_f32(false, a, false, b2, (short)0, acc2, false, false);
        acc3 = __builtin_amdgcn_wmma_f32_16x16x4_f32(false, a, false, b3, (short)0, acc3, false, false);
    }

    const int col0 = tn + c;
    const float bs0 = bias ? bias[col0]      : 0.f;
    const float bs1 = bias ? bias[col0 + 16] : 0.f;
    const float bs2 = bias ? bias[col0 + 32] : 0.f;
    const float bs3 = bias ? bias[col0 + 48] : 0.f;
    const int mBase = tm * 16 + ((lane >> 4) << 3);  // +0 or +8
#pragma unroll
    for (int r = 0; r < 8; ++r) {
        float* Crow = C + (size_t)(mBase + r) * Nout + col0;
        Crow[0]  = acc0[r] + bs0;
        Crow[16] = acc1[r] + bs1;
        Crow[32] = acc2[r] + bs2;
        Crow[48] = acc3[r] + bs3;
    }
}

// ---------------------------------------------------------------------
__device__ __forceinline__ float atomicMaxF(float* addr, float val)
{
    if (val >= 0.f)
        return __int_as_float(atomicMax((int*)addr, __float_as_int(val)));
    else
        return __uint_as_float(atomicMin((unsigned int*)addr, __float_as_uint(val)));
}

__global__ void zero_deg(float* __restrict__ deg)
{
    int i = blockIdx.x * blockDim.x + threadIdx.x;
    if (i < N_NODES) deg[i] = 0.f;
}

__global__ void degree_accum(const int* __restrict__ ei, float* __restrict__ deg)
{
    int e = blockIdx.x * blockDim.x + threadIdx.x;
    if (e < N_EDGES) atomicAdd(&deg[ei[e]], 1.0f);   // source nodes = ei[0, :]
}

// h[n,d] += sin/cos(deg[n] / 10000^(d/D))  (even d -> sin, odd d -> cos)
__global__ void add_degree_pe(const float* __restrict__ deg, float* __restrict__ h)
{
    int n = blockIdx.x;
    int d = threadIdx.x;                 // blockDim.x == 128
    float dg = deg[n];
    float freq = dg * expf(-(float)d * (9.210340371976184f / (float)DM));
    float pe = (d & 1) ? cosf(freq) : sinf(freq);
    h[(size_t)n * DM + d] += pe;
}

__global__ void init_softmax_stats(float* __restrict__ m, float* __restrict__ s)
{
    int i = blockIdx.x * blockDim.x + threadIdx.x;
    if (i < N_NODES * NHEAD) { m[i] = -3.0e38f; s[i] = 0.f; }
}

// Pass A: one wave per (edge, head): score = <q[dst,h,:], k[src,h,:]>/sqrt(D)
__global__ void edge_scores(const int* __restrict__ ei,
                            const float* __restrict__ q,
                            const float* __restrict__ k,
                            float* __restrict__ scores,
                            float* __restrict__ m)
{
    int task = blockIdx.x * (blockDim.x >> 5) + (threadIdx.x >> 5);
    int lane = threadIdx.x & 31;
    if (task >= N_EDGES * NHEAD) return;
    int e  = task >> 2;
    int hh = task & 3;
    int src = ei[e];
    int dst = ei[N_EDGES + e];
    const float4* qv = (const float4*)(q + (size_t)dst * HD + hh * DM);
    const float4* kv = (const float4*)(k + (size_t)src * HD + hh * DM);
    float4 a = qv[lane];
    float4 b = kv[lane];
    float sum = a.x * b.x + a.y * b.y + a.z * b.z + a.w * b.w;
#pragma unroll
    for (int off = 16; off > 0; off >>= 1) sum += __shfl_xor(sum, off, 32);
    if (lane == 0) {
        float sc = sum * 0.08838834764831845f;   // 1/sqrt(128)
        scores[task] = sc;
        atomicMaxF(&m[dst * NHEAD + hh], sc);
    }
}

// Pass B: p = exp(score - m[dst,h]); s[dst,h] += p
__global__ void edge_softmax_num(const int* __restrict__ ei,
                                 float* __restrict__ scores,
                                 const float* __restrict__ m,
                                 float* __restrict__ s)
{
    int idx = blockIdx.x * blockDim.x + threadIdx.x;
    if (idx >= N_EDGES * NHEAD) return;
    int e  = idx >> 2;
    int hh = idx & 3;
    int dst = ei[N_EDGES + e];
    float p = expf(scores[idx] - m[dst * NHEAD + hh]);
    scores[idx] = p;
    atomicAdd(&s[dst * NHEAD + hh], p);
}

// Pass C: h2[dst,d] += (1/H) * sum_h alpha(e,h) * v[src,h,d]
__global__ void edge_aggregate(const int* __restrict__ ei,
                               const float* __restrict__ scores,
                               const float* __restrict__ s,
                               const float* __restrict__ v,
                               float* __restrict__ h2)
{
    int e = blockIdx.x;
    int d = threadIdx.x;                 // blockDim.x == 128
    int src = ei[e];
    int dst = ei[N_EDGES + e];
    float acc = 0.f;
#pragma unroll
    for (int hh = 0; hh < NHEAD; ++hh) {
        float alpha = scores[e * NHEAD + hh] / (s[dst * NHEAD + hh] + 1e-16f);
        acc += alpha * v[(size_t)src * HD + hh * DM + d];
    }
    atomicAdd(&h2[(size_t)dst * DM + d], acc * 0.25f);
}

// h = LN(h + h2) * g + b  (optional ReLU).  One wave per node, 4 elems/lane.
__global__ void add_ln_relu(const float* __restrict__ hin,
                            const float* __restrict__ h2,
                            const float* __restrict__ g,
                            const float* __restrict__ b,
                            float* __restrict__ hout,
                            int doRelu)
{
    int node = blockIdx.x * (blockDim.x >> 5) + (threadIdx.x >> 5);
    int lane = threadIdx.x & 31;
    if (node >= N_NODES) return;
    float4 ya = ((const float4*)(hin + (size_t)node * DM))[lane];
    float4 yb = ((const float4*)(h2  + (size_t)node * DM))[lane];
    float yx = ya.x + yb.x, yy = ya.y + yb.y, yz = ya.z + yb.z, yw = ya.w + yb.w;
    float sum = yx + yy + yz + yw;
#pragma unroll
    for (int off = 16; off > 0; off >>= 1) sum += __shfl_xor(sum, off, 32);
    float mu = sum * (1.0f / DM);
    float dx = yx - mu, dy = yy - mu, dz = yz - mu, dw = yw - mu;
    float vs = dx * dx + dy * dy + dz * dz + dw * dw;
#pragma unroll
    for (int off = 16; off > 0; off >>= 1) vs += __shfl_xor(vs, off, 32);
    float rstd = rsqrtf(vs * (1.0f / DM) + LN_EPS);
    float4 gg = ((const float4*)g)[lane];
    float4 bb = ((const float4*)b)[lane];
    float4 o;
    o.x = dx * rstd * gg.x + bb.x;
    o.y = dy * rstd * gg.y + bb.y;
    o.z = dz * rstd * gg.z + bb.z;
    o.w = dw * rstd * gg.w + bb.w;
    if (doRelu) {
        o.x = fmaxf(o.x, 0.f); o.y = fmaxf(o.y, 0.f);
        o.z = fmaxf(o.z, 0.f); o.w = fmaxf(o.w, 0.f);
    }
    ((float4*)(hout + (size_t)node * DM))[lane] = o;
}

// out[n,c] = h[n,:] @ Wc[:,c] + bc[c]   (tiny GEMV, one thread per (n,c))
__global__ void classifier(const float* __restrict__ h,
                           const float* __restrict__ Wc,
                           const float* __restrict__ bc,
                           float* __restrict__ out)
{
    int t = blockIdx.x * blockDim.x + threadIdx.x;
    int n = t >> 3;
    int c = t & 7;
    if (n >= N_NODES || c >= NUM_CLASSES) return;
    float acc = bc[c];
#pragma unroll 4
    for (int d = 0; d < DM; ++d)
        acc += h[(size_t)n * DM + d] * Wc[d * NUM_CLASSES + c];
    out[n * NUM_CLASSES + c] = acc;
}

// =====================================================================
static inline void launch_gemm(const float* A, const float* B, const float* bias,
                               float* C, int M, int K, int Nout, hipStream_t stream)
{
    int nGroups = Nout >> 6;
    int Mtiles  = (M + 15) >> 4;
    int rowBlks = (Mtiles + 7) >> 3;            // 8 waves (8 row tiles) per block
    // padded LDS row stride if it fits in the 64 KB dynamic-LDS budget
    int bstride = ((size_t)K * 65 * 4 <= 65536) ? 65 : 64;
    size_t shmem = (size_t)K * bstride * 4;
    gemm_wmma_f32<<<rowBlks * nGroups, 256, shmem, stream>>>(A, B, bias, C, M, K, Nout, bstride);
}

extern "C" void kernel_launch(void* const* d_in, const int* in_sizes, int n_in,
                              void* d_out, int out_size, void* d_ws, size_t ws_size,
                              hipStream_t stream)
{
    const float* x     = (const float*)d_in[0];
    const int*   ei    = (const int*)  d_in[1];
    const float* W_in  = (const float*)d_in[2];
    const float* b_in  = (const float*)d_in[3];
    const float* Wq    = (const float*)d_in[4];
    const float* bq    = (const float*)d_in[5];
    const float* Wk    = (const float*)d_in[6];
    const float* bk    = (const float*)d_in[7];
    const float* Wv    = (const float*)d_in[8];
    const float* bv    = (const float*)d_in[9];
    const float* Wsk   = (const float*)d_in[10];
    const float* bsk   = (const float*)d_in[11];
    const float* ln_g  = (const float*)d_in[12];
    const float* ln_b  = (const float*)d_in[13];
    const float* Wc    = (const float*)d_in[14];
    const float* bc    = (const float*)d_in[15];
    float* out = (float*)d_out;

    // workspace carve-up (~150 MB of f32)
    float* ws = (float*)d_ws;
    size_t o = 0;
    float* h      = ws + o; o += (size_t)N_NODES * DM;
    float* h2     = ws + o; o += (size_t)N_NODES * DM;
    float* q      = ws + o; o += (size_t)N_NODES * HD;
    float* kbuf   = ws + o; o += (size_t)N_NODES * HD;
    float* vbuf   = ws + o; o += (size_t)N_NODES * HD;
    float* scores = ws + o; o += (size_t)N_EDGES * NHEAD;
    float* mbuf   = ws + o; o += (size_t)N_NODES * NHEAD;
    float* sbuf   = ws + o; o += (size_t)N_NODES * NHEAD;
    float* deg    = ws + o; o += (size_t)N_NODES;

    const int WPB = 8;

    // 1) h = x @ W_in + b_in          [20000 x 256] @ [256 x 128]
    launch_gemm(x, W_in, b_in, h, N_NODES, D_INF, DM, stream);

    // 2) degree positional encoding
    zero_deg<<<(N_NODES + 255) / 256, 256, 0, stream>>>(deg);
    degree_accum<<<(N_EDGES + 255) / 256, 256, 0, stream>>>(ei, deg);
    add_degree_pe<<<N_NODES, DM, 0, stream>>>(deg, h);

    // 3) transformer layers
    for (int l = 0; l < NLAYER; ++l) {
        const float* Wq_l  = Wq  + (size_t)l * DM * HD;
        const float* bq_l  = bq  + (size_t)l * HD;
        const float* Wk_l  = Wk  + (size_t)l * DM * HD;
        const float* bk_l  = bk  + (size_t)l * HD;
        const float* Wv_l  = Wv  + (size_t)l * DM * HD;
        const float* bv_l  = bv  + (size_t)l * HD;
        const float* Wsk_l = Wsk + (size_t)l * DM * DM;
        const float* bsk_l = bsk + (size_t)l * DM;

        init_softmax_stats<<<(N_NODES * NHEAD + 255) / 256, 256, 0, stream>>>(mbuf, sbuf);

        // q/k/v: [20000 x 128] @ [128 x 512]
        launch_gemm(h, Wq_l, bq_l, q,    N_NODES, DM, HD, stream);
        launch_gemm(h, Wk_l, bk_l, kbuf, N_NODES, DM, HD, stream);
        launch_gemm(h, Wv_l, bv_l, vbuf, N_NODES, DM, HD, stream);

        // skip: h2 = h @ Wskip + bskip  (attention agg atomically added on top)
        launch_gemm(h, Wsk_l, bsk_l, h2, N_NODES, DM, DM, stream);

        // edge softmax attention
        {
            int tasks  = N_EDGES * NHEAD;                   // one wave each
            int blocks = (tasks + WPB - 1) / WPB;
            edge_scores<<<blocks, 256, 0, stream>>>(ei, q, kbuf, scores, mbuf);
        }
        edge_softmax_num<<<(N_EDGES * NHEAD + 255) / 256, 256, 0, stream>>>(ei, scores, mbuf, sbuf);
        edge_aggregate<<<N_EDGES, DM, 0, stream>>>(ei, scores, sbuf, vbuf, h2);

        // h = LN(h + h2) (+ReLU except last layer); wave-per-node, in place
        add_ln_relu<<<(N_NODES + WPB - 1) / WPB, 256, 0, stream>>>(
            h, h2, ln_g + (size_t)l * DM, ln_b + (size_t)l * DM, h, (l < NLAYER - 1) ? 1 : 0);
    }

    // 4) classifier
    classifier<<<(N_NODES * 8 + 255) / 256, 256, 0, stream>>>(h, Wc, bc, out);
}